// Net_46858093199676
// MI455X (gfx1250) — compile-verified
//
#include <hip/hip_runtime.h>

#define NN 50000
#define LD 64

typedef __attribute__((ext_vector_type(2))) float v2f;
typedef __attribute__((ext_vector_type(8))) float v8f;

// ---------------- degree / norm ----------------
__global__ void init_deg_k(float* deg, int n) {
  int i = blockIdx.x * blockDim.x + threadIdx.x;
  if (i < n) deg[i] = 1.0f;  // self-loop contribution
}

__global__ void deg_atomic_k(const int* __restrict__ col, float* deg, int E) {
  int e = blockIdx.x * blockDim.x + threadIdx.x;
  if (e < E) atomicAdd(&deg[col[e]], 1.0f);
}

__global__ void dinv_k(const float* __restrict__ deg, float* __restrict__ dinv, int n) {
  int i = blockIdx.x * blockDim.x + threadIdx.x;
  if (i < n) dinv[i] = rsqrtf(deg[i]);
}

// ---------------- padding ----------------
__global__ void pad_w_k(const float* __restrict__ W, float* __restrict__ Wp, int K, int N) {
  int tid = blockIdx.x * blockDim.x + threadIdx.x;  // 64*64
  if (tid >= 64 * 64) return;
  int k = tid >> 6, n = tid & 63;
  Wp[tid] = (k < K && n < N) ? W[k * N + n] : 0.0f;
}

__global__ void pad_x_k(const float* __restrict__ x, float* __restrict__ xp, int n) {
  int tid = blockIdx.x * blockDim.x + threadIdx.x;  // n*64
  if (tid >= n * LD) return;
  int i = tid >> 6, c = tid & 63;
  xp[tid] = (c < 50) ? x[i * 50 + c] : 0.0f;
}

// ---------------- WMMA f32 GEMM: out[r,c] = dinv[r] * sum_k A[r,k]*Bp[k,c] ----------------
// A: [M x 64] padded (cols >= Kvalid are zero), Bp: [64 x 64] padded, out: [M x 64]
// grid.x = M/16 tiles, 128 threads = 4 waves, wave w owns N-tile w (cols 16w..16w+15)
__global__ void __launch_bounds__(128) gemm_wmma_k(const float* __restrict__ A,
                                                   const float* __restrict__ Bp,
                                                   const float* __restrict__ dinv,
                                                   float* __restrict__ out) {
  const int lane = threadIdx.x & 31;
  const int wave = threadIdx.x >> 5;  // N-tile 0..3
  const int half = lane >> 4;         // 0: K pair {0,1}; 1: K pair {2,3}
  const int l = lane & 15;
  const int rowA = blockIdx.x * 16 + l;  // A row for this lane (A striped in-lane)
  const int colB = wave * 16 + l;        // B col for this lane (B striped across lanes)
  const float* __restrict__ arow = A + (size_t)rowA * LD;

  v8f acc = {};
#pragma unroll
  for (int kb = 0; kb < 13; ++kb) {  // K = 52 >= 50, padded region is zero
    const int k0 = kb * 4 + half * 2;
    v2f a, b;
    a.x = arow[k0];
    a.y = arow[k0 + 1];
    b.x = Bp[k0 * LD + colB];
    b.y = Bp[(k0 + 1) * LD + colB];
    acc = __builtin_amdgcn_wmma_f32_16x16x4_f32(
        /*neg_a=*/false, a, /*neg_b=*/false, b,
        /*c_mod=*/(short)0, acc, /*reuse_a=*/false, /*reuse_b=*/false);
  }

  // D layout: VGPR v -> M = v + 8*half, N = lane&15 (within 16x16 tile)
  const int cout = wave * 16 + l;
#pragma unroll
  for (int v = 0; v < 8; ++v) {
    const int r = blockIdx.x * 16 + half * 8 + v;
    out[(size_t)r * LD + cout] = acc[v] * dinv[r];
  }
}

// ---------------- edge scatter: acc[t,:] += p[s,:] ----------------
__global__ void scatter_k(const float* __restrict__ p, float* __restrict__ acc,
                          const int* __restrict__ rows, const int* __restrict__ cols, int E) {
  int tid = blockIdx.x * blockDim.x + threadIdx.x;  // E*64 < 2^31
  if (tid >= E * 64) return;
  const int e = tid >> 6;
  const int c = tid & 63;
  const int s = rows[e];
  const int t = cols[e];
  const float val = p[(size_t)s * LD + c];
  if (val != 0.0f) atomicAdd(&acc[(size_t)t * LD + c], val);
}

// ---------------- finalize ----------------
__global__ void finalize_relu_k(const float* __restrict__ acc, const float* __restrict__ dinv,
                                const float* __restrict__ bias, float* __restrict__ h, int n) {
  int tid = blockIdx.x * blockDim.x + threadIdx.x;
  if (tid >= n * LD) return;
  const int i = tid >> 6, c = tid & 63;
  float v = 0.0f;
  if (c < 50) v = fmaxf(dinv[i] * acc[tid] + bias[c], 0.0f);
  h[tid] = v;  // cols 50..63 zeroed so next GEMM needs no K guard
}

__global__ void finalize_z_k(const float* __restrict__ acc, const float* __restrict__ dinv,
                             const float* __restrict__ bias, float* __restrict__ z, int n) {
  int tid = blockIdx.x * blockDim.x + threadIdx.x;
  if (tid >= n * LD) return;
  const int i = tid >> 6, c = tid & 63;
  z[tid] = dinv[i] * acc[tid] + bias[c];
}

// ---------------- decode: logits[e] = dot(z[a], z[b]) over 64 dims ----------------
__global__ void decode_k(const float* __restrict__ z, const int* __restrict__ pos,
                         const int* __restrict__ neg, float* __restrict__ out, int Ep, int En) {
  int e = blockIdx.x * blockDim.x + threadIdx.x;
  if (e >= Ep + En) return;
  int a, b;
  if (e < Ep) {
    a = pos[e];
    b = pos[Ep + e];
  } else {
    const int f = e - Ep;
    a = neg[f];
    b = neg[En + f];
  }
  const float4* __restrict__ za = (const float4*)(z + (size_t)a * LD);
  const float4* __restrict__ zb = (const float4*)(z + (size_t)b * LD);
  float s = 0.0f;
#pragma unroll
  for (int i = 0; i < 16; ++i) {
    const float4 u = za[i];
    const float4 v = zb[i];
    s += u.x * v.x + u.y * v.y + u.z * v.z + u.w * v.w;
  }
  out[e] = s;
}

extern "C" void kernel_launch(void* const* d_in, const int* in_sizes, int n_in,
                              void* d_out, int out_size, void* d_ws, size_t ws_size,
                              hipStream_t stream) {
  const float* x  = (const float*)d_in[0];
  const int* tr   = (const int*)d_in[1];
  const int* pos  = (const int*)d_in[2];
  const int* neg  = (const int*)d_in[3];
  const float* W1 = (const float*)d_in[4];
  const float* b1 = (const float*)d_in[5];
  const float* W2 = (const float*)d_in[6];
  const float* b2 = (const float*)d_in[7];
  float* out = (float*)d_out;

  const int E  = in_sizes[1] / 2;
  const int Ep = in_sizes[2] / 2;
  const int En = in_sizes[3] / 2;
  const int N  = NN;

  // workspace carve-out (all 16B-aligned offsets)
  float* w = (float*)d_ws;
  size_t o = 0;
  float* deg  = w + o; o += 50048;
  float* dinv = w + o; o += 50048;
  float* Wp1  = w + o; o += 4096;
  float* Wp2  = w + o; o += 4096;
  float* bufA = w + o; o += (size_t)N * LD;
  float* bufB = w + o; o += (size_t)N * LD;
  float* bufC = w + o; o += (size_t)N * LD;
  (void)ws_size; (void)n_in; (void)out_size;

  const int* trRow = tr;      // edge_index[0] (sources)
  const int* trCol = tr + E;  // edge_index[1] (targets)

  const int nodeElems = N * LD;
  const size_t featBytes = (size_t)N * LD * sizeof(float);
  const int scatterBlocks = (E * 64 + 255) / 256;

  // 1) degree + dinv (self-loops folded into init)
  init_deg_k<<<(N + 255) / 256, 256, 0, stream>>>(deg, N);
  deg_atomic_k<<<(E + 255) / 256, 256, 0, stream>>>(trCol, deg, E);
  dinv_k<<<(N + 255) / 256, 256, 0, stream>>>(deg, dinv, N);

  // 2) pad weights and x
  pad_w_k<<<16, 256, 0, stream>>>(W1, Wp1, 50, 50);
  pad_w_k<<<16, 256, 0, stream>>>(W2, Wp2, 50, 64);
  pad_x_k<<<(nodeElems + 255) / 256, 256, 0, stream>>>(x, bufC, N);

  // 3) layer 1: p1 = dinv .* (xpad @ Wp1)   [bufC -> bufA]
  gemm_wmma_k<<<N / 16, 128, 0, stream>>>(bufC, Wp1, dinv, bufA);
  hipMemcpyAsync(bufB, bufA, featBytes, hipMemcpyDeviceToDevice, stream);  // self-loop seed
  scatter_k<<<scatterBlocks, 256, 0, stream>>>(bufA, bufB, trRow, trCol, E);
  finalize_relu_k<<<(nodeElems + 255) / 256, 256, 0, stream>>>(bufB, dinv, b1, bufA, N);

  // 4) layer 2: p2 = dinv .* (h @ Wp2)      [bufA -> bufB]
  gemm_wmma_k<<<N / 16, 128, 0, stream>>>(bufA, Wp2, dinv, bufB);
  hipMemcpyAsync(bufC, bufB, featBytes, hipMemcpyDeviceToDevice, stream);  // self-loop seed
  scatter_k<<<scatterBlocks, 256, 0, stream>>>(bufB, bufC, trRow, trCol, E);
  finalize_z_k<<<(nodeElems + 255) / 256, 256, 0, stream>>>(bufC, dinv, b2, bufA, N);

  // 5) decode
  decode_k<<<(Ep + En + 255) / 256, 256, 0, stream>>>(bufA, pos, neg, out, Ep, En);
}